// Quantum1DCovolution_73435350827496
// MI455X (gfx1250) — compile-verified
//
#include <hip/hip_runtime.h>

typedef float v2f __attribute__((ext_vector_type(2)));
typedef float v8f __attribute__((ext_vector_type(8)));

#define DIM 256          // 2^8 statevector size
#define WREAL 512        // real-representation dimension
#define BTILE 16         // batch items per block in GEMM kernel
#define SU_STRIDE 516    // padded LDS row stride in floats (516%64==4 -> conflict-free A reads)

// ---------------------------------------------------------------------------
// Kernel A: build the circuit unitary U (depends only on weights) and store
// its real 512x512 representation W = [[Ur,-Ui],[Ui,Ur]] in a WMMA
// B-fragment-packed layout:
//   for k-step ks=k/4, h=(k>>1)&1, lo=k&1, column n:
//     wsB[ks*2048 + n*4 + h*2 + lo] = W[n][k]   (i.e. W transposed, packed)
// so that in the GEMM, lane l of a wave loads its (VGPR0,VGPR1)=(K=k0+2h,
// K=k0+2h+1) B pair as one contiguous float2 (perfectly coalesced: 32 lanes
// cover exactly 256 contiguous bytes per 16-column tile per k-step).
// One block per basis column j; 256 threads each own one amplitude.
// ---------------------------------------------------------------------------
__global__ __launch_bounds__(256)
void qcnn_build_unitary(const float* __restrict__ w, float* __restrict__ wsB)
{
    __shared__ float sre[DIM];
    __shared__ float sim[DIM];
    const int i = threadIdx.x;   // amplitude index (bit 7 = qubit 0 ... bit 0 = qubit 7)
    const int j = blockIdx.x;    // basis column

    sre[i] = (i == j) ? 1.0f : 0.0f;
    sim[i] = 0.0f;
    __syncthreads();

    const int pa[7] = {0, 2, 4, 6, 0, 4, 0};
    const int pb[7] = {1, 3, 5, 7, 2, 6, 4};
    const float cY = 0.9238795325112867f;   // cos(pi/8)
    const float sY = 0.3826834323650898f;   // sin(pi/8)

#pragma unroll
    for (int k = 0; k < 7; ++k) {
        const int cb = 7 - pa[k];          // bit position of qubit a
        const int tb = 7 - pb[k];          // bit position of qubit b
        const float phi0 = w[2 * k + 0];
        const float phi1 = w[2 * k + 1];

        // ---- CRZ(phi0), control=a, target=b (diagonal) ----
        {
            float re = sre[i], im = sim[i];
            if ((i >> cb) & 1) {
                float ph = ((i >> tb) & 1) ? 0.5f * phi0 : -0.5f * phi0;  // amp *= e^{i*ph}
                float c = cosf(ph), s = sinf(ph);
                sre[i] = re * c - im * s;
                sim[i] = re * s + im * c;
            }
        }
        __syncthreads();

        // ---- CRX(phi1), control=a, target=b ----
        {
            int p = i ^ (1 << tb);
            float mre = sre[i], mim = sim[i];
            float pre = sre[p], pim = sim[p];
            __syncthreads();
            if ((i >> cb) & 1) {
                float c = cosf(0.5f * phi1), s = sinf(0.5f * phi1);
                // new = c*me + (-i s)*partner
                sre[i] = c * mre + s * pim;
                sim[i] = c * mim - s * pre;
            }
            __syncthreads();
        }

        // ---- CRY(pi/4), control=b, target=a ----
        {
            int p = i ^ (1 << cb);
            float mre = sre[i], mim = sim[i];
            float pre = sre[p], pim = sim[p];
            __syncthreads();
            if ((i >> tb) & 1) {
                if (((i >> cb) & 1) == 0) {     // my target bit = 0
                    sre[i] = cY * mre - sY * pre;
                    sim[i] = cY * mim - sY * pim;
                } else {                        // my target bit = 1
                    sre[i] = cY * mre + sY * pre;
                    sim[i] = cY * mim + sY * pim;
                }
            }
            __syncthreads();
        }
    }

    // U[i][j] = sre[i] + i*sim[i]. Emit 4 entries of W^T in packed layout.
    const float re = sre[i];
    const float im = sim[i];
    auto put = [&](int kk, int nn, float v) {
        int ks = kk >> 2, h = (kk >> 1) & 1, lo = kk & 1;
        wsB[ks * 2048 + nn * 4 + h * 2 + lo] = v;
    };
    put(j,       i,       re);    // W[i][j]           = Ur
    put(j,       i + 256, im);    // W[i+256][j]       = Ui
    put(j + 256, i,      -im);    // W[i][j+256]       = -Ui
    put(j + 256, i + 256, re);    // W[i+256][j+256]   = Ur
}

// ---------------------------------------------------------------------------
// Kernel B: per block handle 16 batch items.
//  Phase 1: build u_b = [Re psi_b ; Im psi_b] (product state) into LDS.
//           amp_i = (prod_q bit?sin:cos(x_q/2)) * (-i)^popcount(i)
//  Phase 2: Psi' = U_batch * W^T via V_WMMA_F32_16X16X4_F32, K=512.
//           8 waves x 4 col-tiles each = 32 tiles of 16x16.
//  Epilogue: E_b = sum_n sign(n) * Psi'[b][n]^2,  sign = +1 if (n&255)<128.
// ---------------------------------------------------------------------------
__global__ __launch_bounds__(256)
void qcnn_expect(const float* __restrict__ x,
                 const float* __restrict__ wsB,
                 float* __restrict__ out)
{
    __shared__ float su[BTILE * SU_STRIDE];   // 16 x 512 (padded) = ~33 KB
    __shared__ float eacc[BTILE];

    const int t    = threadIdx.x;
    const int lane = t & 31;
    const int wave = t >> 5;          // 0..7
    const int b0   = blockIdx.x * BTILE;

    if (t < BTILE) eacc[t] = 0.0f;

    // ---------------- Phase 1: product states ----------------
    {
        const int item = t >> 4;      // 0..15
        const int sub  = t & 15;      // 16 threads per item, 16 amps each
        float c[8], s[8];
#pragma unroll
        for (int q = 0; q < 8; ++q) {
            float xv = x[(size_t)(b0 + item) * 8 + q] * 0.5f;
            c[q] = cosf(xv);
            s[q] = sinf(xv);
        }
        const float phr[4] = {1.0f, 0.0f, -1.0f, 0.0f};   // Re (-i)^pc
        const float phi[4] = {0.0f, -1.0f, 0.0f, 1.0f};   // Im (-i)^pc
#pragma unroll
        for (int a = 0; a < 16; ++a) {
            int idx = sub * 16 + a;
            float m = 1.0f;
#pragma unroll
            for (int q = 0; q < 8; ++q)
                m *= ((idx >> (7 - q)) & 1) ? s[q] : c[q];
            int pc = __popc(idx) & 3;
            su[item * SU_STRIDE + idx]       = m * phr[pc];   // p_idx
            su[item * SU_STRIDE + 256 + idx] = m * phi[pc];   // q_idx
        }
    }
    __syncthreads();

    // ---------------- Phase 2: WMMA GEMM ----------------
    const int ctbase = wave * 4;          // this wave's 4 column tiles
    const int h      = lane >> 4;         // 0/1 half-wave (K split)
    const int mi     = lane & 15;         // M (batch row) / N (col) within tile
    const int item   = mi;

    v8f acc[4] = {};                      // 4 tiles of 16x16 f32 accumulators

    const float* aBase = &su[item * SU_STRIDE + 2 * h];
    for (int ks = 0; ks < 128; ++ks) {
        const int k0 = ks * 4;
        v2f a;                            // A: VGPR0=K(k0+2h), VGPR1=K(k0+2h+1)
        a.x = aBase[k0];
        a.y = aBase[k0 + 1];

        const float* bRow = wsB + (size_t)ks * 2048 + mi * 4 + h * 2;
        int ksp = (ks + 4 < 128) ? (ks + 4) : ks;
        __builtin_prefetch(wsB + (size_t)ksp * 2048 + ctbase * 64 + lane * 8, 0, 0);

#pragma unroll
        for (int tt = 0; tt < 4; ++tt) {
            const float* bp = bRow + (ctbase + tt) * 64;   // +n_tile*16 cols *4
            v2f b;
            b.x = bp[0];
            b.y = bp[1];
            acc[tt] = __builtin_amdgcn_wmma_f32_16x16x4_f32(
                false, a, false, b, (short)0, acc[tt], false, false);
        }
    }

    // ---------------- Epilogue: signed squared reduction ----------------
    float part[8];
#pragma unroll
    for (int v = 0; v < 8; ++v) part[v] = 0.0f;
#pragma unroll
    for (int tt = 0; tt < 4; ++tt) {
        int n = (ctbase + tt) * 16 + mi;                 // column in [0,512)
        float sgn = ((n & 255) < 128) ? 1.0f : -1.0f;    // Z on qubit 0
#pragma unroll
        for (int v = 0; v < 8; ++v) {
            float d = acc[tt][v];                        // element (M=h*8+v, N=mi)
            part[v] += sgn * d * d;
        }
    }
#pragma unroll
    for (int v = 0; v < 8; ++v)
        atomicAdd(&eacc[h * 8 + v], part[v]);            // ds_add_f32

    __syncthreads();
    if (t < BTILE) out[b0 + t] = eacc[t];
}

// ---------------------------------------------------------------------------
extern "C" void kernel_launch(void* const* d_in, const int* in_sizes, int n_in,
                              void* d_out, int out_size, void* d_ws, size_t ws_size,
                              hipStream_t stream) {
    const float* x = (const float*)d_in[0];   // (65536, 8) f32
    const float* w = (const float*)d_in[1];   // (7, 2) f32
    float* out = (float*)d_out;               // (65536,) f32
    float* wsB = (float*)d_ws;                // needs 512*512*4 = 1 MB scratch

    // Build batch-invariant circuit unitary (packed B-fragment layout).
    qcnn_build_unitary<<<DIM, 256, 0, stream>>>(w, wsB);

    // Batched expectation via WMMA GEMM.
    const int batch = in_sizes[0] / 8;
    qcnn_expect<<<batch / BTILE, 256, 0, stream>>>(x, wsB, out);
}